// RQCNN_68951404970219
// MI455X (gfx1250) — compile-verified
//
#include <hip/hip_runtime.h>
#include <math.h>

// Problem constants (match reference)
#define NQ_     4
#define LAYERS_ 4
#define B_      32
#define C_      3
#define H_      128
#define W_      128
#define R_      127            // H - K + 1

typedef __attribute__((ext_vector_type(2))) float v2f;
typedef __attribute__((ext_vector_type(8))) float v8f;

// ---------------------------------------------------------------------------
// Setup kernel (1 block, 16 threads): thread j evolves basis state e_j through
// the 4-layer circuit, producing column j of the 16x16 real unitary U.
// Writes Uws[col*16 + row] = U[row][col] to workspace.
// ---------------------------------------------------------------------------
__global__ void quanv_build_u_kernel(const float* __restrict__ wts,
                                     float* __restrict__ Uws) {
  const int col = threadIdx.x;
  if (col >= 16) return;

  float st[16];
#pragma unroll
  for (int i = 0; i < 16; ++i) st[i] = (i == col) ? 1.0f : 0.0f;

  for (int l = 0; l < LAYERS_; ++l) {
    // RY(w[l][q]) on qubit q (qubit q <-> index stride 8>>q)
#pragma unroll
    for (int q = 0; q < 4; ++q) {
      float h = 0.5f * wts[l * NQ_ + q];     // in [0, pi): fast-path safe
      float c = __cosf(h), s = __sinf(h);
      int str = 8 >> q;
#pragma unroll
      for (int i = 0; i < 16; ++i) {
        if ((i & str) == 0) {
          float a0 = st[i], a1 = st[i + str];
          st[i]       = c * a0 - s * a1;
          st[i + str] = s * a0 + c * a1;
        }
      }
    }
    // CNOT(0,1), CNOT(2,3), CNOT(1,2): swap target bit on ctrl==1 slab
    const int cs[3] = {8, 2, 4}, ts[3] = {4, 1, 2};
#pragma unroll
    for (int g = 0; g < 3; ++g) {
#pragma unroll
      for (int i = 0; i < 16; ++i) {
        if ((i & cs[g]) && !(i & ts[g])) {
          float t = st[i]; st[i] = st[i + ts[g]]; st[i + ts[g]] = t;
        }
      }
    }
  }
#pragma unroll
  for (int i = 0; i < 16; ++i) Uws[col * 16 + i] = st[i];
}

// ---------------------------------------------------------------------------
// Main kernel: one wave handles 16 output positions (b,i,j).
//   Y[16x16] = S[16 patches x 16 states] x U^T  via 4 chained
//   V_WMMA_F32_16X16X4_F32 per channel; psum += Y^2 over the 3 channels;
//   a 4-stage shfl_xor Walsh-Hadamard then yields all 4 Z expectations,
//   landing on lanes 8/4/2/1 of each 16-lane half, which store directly.
// ---------------------------------------------------------------------------
__global__ __launch_bounds__(256) void quanv_wmma_kernel(
    const float* __restrict__ x,    // [B,C,H,W]
    const float* __restrict__ Uws,  // [16,16] column-major (Uws[col*16+row])
    float* __restrict__ out)        // [B,R,R,4] flat (== reshape(B,4,R,R))
{
  // Ulds[col*16 + row] = U[row][col]
  __shared__ float Ulds[256];
  const int tid = threadIdx.x;
  Ulds[tid] = Uws[tid];
  __syncthreads();

  const int lane = tid & 31;
  const int nSub = lane & 15;     // M-row slot (A) / N-col slot (B,D)
  const bool hi  = lane >= 16;

  // B fragments: B_k[K][N] = U[N][4k+K]; mirror of A layout:
  // lanes 0-15 carry K rows {0,1}, lanes 16-31 carry {2,3}.
  v2f bfrag[4];
#pragma unroll
  for (int k = 0; k < 4; ++k) {
    int i0 = 4 * k + (hi ? 2 : 0);
    bfrag[k].x = Ulds[(i0    ) * 16 + nSub];
    bfrag[k].y = Ulds[(i0 + 1) * 16 + nSub];
  }

  const int Pout = B_ * R_ * R_;              // 516128 output positions
  const int tile = blockIdx.x * (blockDim.x >> 5) + (tid >> 5);
  const int pos  = tile * 16 + nSub;
  const int posc = (pos < Pout) ? pos : (Pout - 1);   // clamp for loads

  int b   = posc / (R_ * R_);
  int rem = posc - b * (R_ * R_);
  int ii  = rem / R_;
  int jj  = rem - ii * R_;

  const float HPI = 1.57079632679489662f;     // pi/2

  v8f psum = {};
#pragma unroll
  for (int c = 0; c < C_; ++c) {
    const float* px = x + (((size_t)b * C_ + c) * H_ + ii) * W_ + jj;
    float p0 = px[0], p1 = px[1], p2 = px[W_], p3 = px[W_ + 1];

    // RY(pi*x)/2 encoding amplitudes; args in [0, pi/2]: fast-path safe
    float amp0[2] = {__cosf(HPI * p0), __sinf(HPI * p0)};
    float amp1[2] = {__cosf(HPI * p1), __sinf(HPI * p1)};
    float amp2[2] = {__cosf(HPI * p2), __sinf(HPI * p2)};
    float amp3[2] = {__cosf(HPI * p3), __sinf(HPI * p3)};

    // Product state s[a*8+b*4+c*2+d]
    float s[16];
#pragma unroll
    for (int a = 0; a < 2; ++a)
#pragma unroll
      for (int bq = 0; bq < 2; ++bq) {
        float ab = amp0[a] * amp1[bq];
#pragma unroll
        for (int cq = 0; cq < 2; ++cq) {
          float abc = ab * amp2[cq];
          s[a * 8 + bq * 4 + cq * 2 + 0] = abc * amp3[0];
          s[a * 8 + bq * 4 + cq * 2 + 1] = abc * amp3[1];
        }
      }

    // Y = S x U^T : 4 chained f32 WMMAs over K=16
    v8f acc = {};
#pragma unroll
    for (int k = 0; k < 4; ++k) {
      v2f afrag;
      afrag.x = hi ? s[4 * k + 2] : s[4 * k + 0];
      afrag.y = hi ? s[4 * k + 3] : s[4 * k + 1];
      acc = __builtin_amdgcn_wmma_f32_16x16x4_f32(
          false, afrag, false, bfrag[k], (short)0, acc, false, false);
    }

    // Channel-summed probabilities
#pragma unroll
    for (int r = 0; r < 8; ++r) psum[r] += acc[r] * acc[r];
  }

  // Z expectations: partial Walsh-Hadamard per row. After 4 butterfly
  // stages, lane h (per 16-lane half) holds
  //   sum_n psum[M][n] * (-1)^popcount(h & n);
  // z_q needs the sign on bit (3-q) only -> h = 8,4,2,1 for q = 0,1,2,3.
#pragma unroll
  for (int r = 0; r < 8; ++r) {
    float v = psum[r];
#pragma unroll
    for (int m = 1; m <= 8; m <<= 1) {
      float y = __shfl_xor(v, m, 32);
      v = (lane & m) ? (y - v) : (v + y);
    }
    int M  = r + (hi ? 8 : 0);
    int p2 = tile * 16 + M;
    int q  = (nSub == 8) ? 0 : (nSub == 4) ? 1 : (nSub == 2) ? 2
                                           : (nSub == 1) ? 3 : -1;
    if (q >= 0 && p2 < Pout) {
      out[(size_t)p2 * 4 + q] = v;   // [B,R,R,4] flat == ref's reshape
    }
  }
}

extern "C" void kernel_launch(void* const* d_in, const int* in_sizes, int n_in,
                              void* d_out, int out_size, void* d_ws, size_t ws_size,
                              hipStream_t stream) {
  const float* x   = (const float*)d_in[0];   // [32,3,128,128] f32
  const float* wts = (const float*)d_in[1];   // [4,4] f32
  float* out = (float*)d_out;                 // [32,4,127,127] f32
  float* Uws = (float*)d_ws;                  // 256 floats of scratch

  // Step 1: build the 16x16 circuit matrix once.
  quanv_build_u_kernel<<<1, 16, 0, stream>>>(wts, Uws);

  // Step 2: quanvolution via f32 WMMA.
  const int Pout = B_ * R_ * R_;
  const int numTiles = (Pout + 15) / 16;      // 16 positions per wave
  const int wavesPerBlock = 8;                // 256 threads = 8 wave32
  const int blocks = (numTiles + wavesPerBlock - 1) / wavesPerBlock;

  quanv_wmma_kernel<<<blocks, 256, 0, stream>>>(x, Uws, out);
}